// msgpass_62723702391420
// MI455X (gfx1250) — compile-verified
//
#include <hip/hip_runtime.h>
#include <stdint.h>

#define FEAT   96
#define WAVE   32
#define BLOCK  256
#define WPB    (BLOCK / WAVE)   // 8 waves per block
#define EPB    512              // edges staged per block (2 KB src + 2 KB dst in LDS)

#if __has_builtin(__builtin_amdgcn_global_load_async_to_lds_b32) && \
    __has_builtin(__builtin_amdgcn_s_wait_asynccnt)
#define USE_ASYNC_LDS 1
#else
#define USE_ASYNC_LDS 0
#endif

typedef __attribute__((address_space(1))) int g_int_t;   // global (AS1) int
typedef __attribute__((address_space(3))) int l_int_t;   // LDS    (AS3) int

__device__ __forceinline__ void atomic_add_f32(float* p, float v) {
    // Native global_atomic_add_f32 (no CAS loop); result not returned -> STOREcnt path.
    unsafeAtomicAdd(p, v);
}

__global__ void msgpass_zero_kernel(float* __restrict__ p, long n) {
    long i      = (long)blockIdx.x * blockDim.x + threadIdx.x;
    long stride = (long)gridDim.x * blockDim.x;
    for (; i < n; i += stride) p[i] = 0.0f;
}

__global__ __launch_bounds__(BLOCK)
void msgpass_scatter_kernel(const float* __restrict__ feature,
                            const int*   __restrict__ src,
                            const int*   __restrict__ dst,
                            float*       __restrict__ outsum,
                            float*       __restrict__ deg,
                            int nE) {
    __shared__ int s_src[EPB];
    __shared__ int s_dst[EPB];

    const int tid    = threadIdx.x;
    const int base   = blockIdx.x * EPB;
    const int nChunk = (nE - base) < EPB ? (nE - base) : EPB;  // block-uniform
    if (nChunk <= 0) return;

    // --- Stage this block's edge indices into LDS via gfx1250 async DMA ---
    for (int i = tid; i < nChunk; i += BLOCK) {
#if USE_ASYNC_LDS
        __builtin_amdgcn_global_load_async_to_lds_b32(
            (g_int_t*)(int*)(src + base + i),
            (l_int_t*)(&s_src[i]), 0, 0);
        __builtin_amdgcn_global_load_async_to_lds_b32(
            (g_int_t*)(int*)(dst + base + i),
            (l_int_t*)(&s_dst[i]), 0, 0);
#else
        s_src[i] = src[base + i];
        s_dst[i] = dst[base + i];
#endif
    }
#if USE_ASYNC_LDS
    __builtin_amdgcn_s_wait_asynccnt(0);
#endif
    __syncthreads();

    const int wave = tid / WAVE;
    const int lane = tid % WAVE;

    // One wave per edge: 3 coalesced 128B loads + 3 coalesced 128B atomic bursts.
    for (int i = wave; i < nChunk; i += WPB) {
        const int s = s_src[i];   // wave-uniform LDS broadcast
        const int d = s_dst[i];
        const float* frow = feature + (long)s * FEAT;
        float*       orow = outsum  + (long)d * FEAT;
        float v0 = frow[lane];
        float v1 = frow[lane + 32];
        float v2 = frow[lane + 64];
        atomic_add_f32(orow + lane,      v0);
        atomic_add_f32(orow + lane + 32, v1);
        atomic_add_f32(orow + lane + 64, v2);
        if (lane == 0) atomic_add_f32(deg + d, 1.0f);
    }
}

__global__ void msgpass_finalize_kernel(float* __restrict__ out,
                                        const float* __restrict__ deg,
                                        int nNodes) {
    const int row = blockIdx.x;
    const int f   = threadIdx.x;
    if (row < nNodes && f < FEAT) {
        float dg    = deg[row];
        float denom = dg > 1.0f ? dg : 1.0f;
        long  idx   = (long)row * FEAT + f;
        out[idx] = out[idx] / denom;   // true divide to match reference
    }
}

extern "C" void kernel_launch(void* const* d_in, const int* in_sizes, int n_in,
                              void* d_out, int out_size, void* d_ws, size_t ws_size,
                              hipStream_t stream) {
    const float* feature = (const float*)d_in[0];
    const int*   src     = (const int*)d_in[1];
    const int*   dst     = (const int*)d_in[2];
    float*       out     = (float*)d_out;
    float*       deg     = (float*)d_ws;     // nNodes floats of scratch

    const int nFeatElems = in_sizes[0];          // nNodes * 96
    const int nNodes     = nFeatElems / FEAT;
    const int nE         = in_sizes[1];

    // Re-zero accumulators every call (harness poisons buffers, replays graph).
    {
        int b1 = (nFeatElems + BLOCK - 1) / BLOCK; if (b1 > 4096) b1 = 4096;
        msgpass_zero_kernel<<<b1, BLOCK, 0, stream>>>(out, (long)nFeatElems);
        int b2 = (nNodes + BLOCK - 1) / BLOCK;
        msgpass_zero_kernel<<<b2, BLOCK, 0, stream>>>(deg, (long)nNodes);
    }

    const int nBlocks = (nE + EPB - 1) / EPB;
    msgpass_scatter_kernel<<<nBlocks, BLOCK, 0, stream>>>(feature, src, dst,
                                                          out, deg, nE);

    msgpass_finalize_kernel<<<nNodes, FEAT, 0, stream>>>(out, deg, nNodes);
}